// ClassWiseECELoss_5634997093213
// MI455X (gfx1250) — compile-verified
//
#include <hip/hip_runtime.h>

// ClassWiseECELoss for MI455X (gfx1250).
// Memory-bound: 200 MB logits @ 23.3 TB/s ~= 8.6 us floor.
// Per-wave 16-row tiles:
//   Phase A1: tile = 1600 contiguous floats -> 12x global_load_b128 + 1x b64,
//             unconditional & coalesced; exp in registers; flat store to LDS.
//   Phase A2: LDS bounce = transpose into WMMA A-fragment order; row sums via
//             V_WMMA_F32_16X16X4_F32 with all-ones B (two accumulators).
//   Phase B : class-striped lanes -> bank-conflict-free ds_add_f32 histogram
//             into block-private (class x bin) conf/acc arrays.
// One global atomic flush per block; tiny finalize emits [sce, per_class[100]].

#define NBINS 15
#define NCLS 100
#define WAVES_PER_BLOCK 8
#define TILE_ROWS 16
#define TILE_ELEMS (TILE_ROWS * NCLS)  // 1600
#define HIST_SEGS (NCLS * NBINS)       // 1500

typedef float v2f __attribute__((ext_vector_type(2)));
typedef float v4f __attribute__((ext_vector_type(4)));
typedef float v8f __attribute__((ext_vector_type(8)));

__global__ void ece_zero_ws(float* __restrict__ ws, int n) {
  int i = blockIdx.x * blockDim.x + threadIdx.x;
  int stride = gridDim.x * blockDim.x;
  for (; i < n; i += stride) ws[i] = 0.0f;
}

__global__ __launch_bounds__(256) void ece_hist_kernel(
    const float* __restrict__ logits, const long long* __restrict__ labels,
    float* __restrict__ g_conf, float* __restrict__ g_acc, int N) {
  __shared__ __align__(16) float elds[WAVES_PER_BLOCK][TILE_ELEMS];  // 51200 B
  __shared__ float rowsum[WAVES_PER_BLOCK][TILE_ROWS];               //   512 B
  __shared__ int lbl_s[WAVES_PER_BLOCK][TILE_ROWS];                  //   512 B
  __shared__ float h_conf[HIST_SEGS];                                //  6000 B
  __shared__ float h_acc[HIST_SEGS];                                 //  6000 B

  const int tid  = threadIdx.x;
  const int w    = tid >> 5;   // wave within block
  const int lane = tid & 31;

  for (int i = tid; i < HIST_SEGS; i += blockDim.x) {
    h_conf[i] = 0.0f;
    h_acc[i]  = 0.0f;
  }
  __syncthreads();

  // WMMA A-matrix (16x4 f32): lanes 0-15 hold K=0,1 of row (lane&15);
  // lanes 16-31 hold K=2,3.
  const int half = lane >> 4;
  const int m    = lane & 15;
  const int totalWaves = gridDim.x * WAVES_PER_BLOCK;
  const int gw = blockIdx.x * WAVES_PER_BLOCK + w;
  const int nTiles = (N + TILE_ROWS - 1) / TILE_ROWS;

  v2f ones;
  ones.x = 1.0f;
  ones.y = 1.0f;

  for (int tile = gw; tile < nTiles; tile += totalWaves) {
    const int rowBase = tile * TILE_ROWS;
    const float* gbase = logits + (long long)rowBase * NCLS;
    float* ew = &elds[w][0];
    const bool fullTile = (rowBase + TILE_ROWS) <= N;

    // ---- Phase A1: coalesced tile load, exp, flat LDS store ---------------
    if (fullTile) {
#pragma unroll
      for (int t = 0; t < 12; ++t) {  // 12 * 128 = 1536 floats
        const int flat = t * 128 + lane * 4;
        v4f x = *(const v4f*)(gbase + flat);
        v4f e;
        e.x = __expf(x.x);
        e.y = __expf(x.y);
        e.z = __expf(x.z);
        e.w = __expf(x.w);
        *(v4f*)(ew + flat) = e;
      }
      {  // tail: 64 floats
        const int flat = 1536 + lane * 2;
        v2f x = *(const v2f*)(gbase + flat);
        v2f e;
        e.x = __expf(x.x);
        e.y = __expf(x.y);
        *(v2f*)(ew + flat) = e;
      }
      if (lane < TILE_ROWS) lbl_s[w][lane] = (int)labels[rowBase + lane];
    } else {
      // Rare ragged tail (never taken for N % 16 == 0): guarded scalar path.
      const long long limit = (long long)N * NCLS;
      for (int t = 0; t < TILE_ELEMS / 32; ++t) {
        const int flat = t * 32 + lane;
        const long long gidx = (long long)rowBase * NCLS + flat;
        ew[flat] = (gidx < limit) ? __expf(gbase[flat]) : 0.0f;
      }
      if (lane < TILE_ROWS)
        lbl_s[w][lane] = (rowBase + lane < N) ? (int)labels[rowBase + lane] : -1;
    }
    // Same-wave LDS ops are in-order: phase A2/B reads see these stores.

    // ---- Phase A2: WMMA row sums from LDS (A fragments, B = ones) ---------
    v8f acc0 = {0.0f, 0.0f, 0.0f, 0.0f, 0.0f, 0.0f, 0.0f, 0.0f};
    v8f acc1 = {0.0f, 0.0f, 0.0f, 0.0f, 0.0f, 0.0f, 0.0f, 0.0f};
    const float* arow = &elds[w][m * NCLS + half * 2];
#pragma unroll
    for (int k0 = 0; k0 < NCLS; k0 += 8) {
      v2f a0 = *(const v2f*)(arow + k0);  // conflict-free ds_load_b64
      acc0 = __builtin_amdgcn_wmma_f32_16x16x4_f32(
          false, a0, false, ones, (short)0, acc0, false, false);
      if (k0 + 4 < NCLS) {
        v2f a1 = *(const v2f*)(arow + k0 + 4);
        acc1 = __builtin_amdgcn_wmma_f32_16x16x4_f32(
            false, a1, false, ones, (short)0, acc1, false, false);
      }
    }
    // D layout: VGPR v -> M=v (lanes 0-15), M=v+8 (lanes 16-31); every column
    // equals the row sum. Lanes 0 and 16 publish rows 0-7 / 8-15.
    if (m == 0) {
      union { v8f v; float f[8]; } u;
      u.v = acc0 + acc1;
#pragma unroll
      for (int v = 0; v < 8; ++v) rowsum[w][half * 8 + v] = u.f[v];
    }

    // ---- Phase B: bin + class-striped histogram atomics -------------------
    for (int r = 0; r < TILE_ROWS; ++r) {
      const int rr = rowBase + r;
      if (rr >= N) break;
      const float inv = 1.0f / rowsum[w][r];
      const int lbl = lbl_s[w][r];
      const float* er = &elds[w][r * NCLS];
#pragma unroll
      for (int j = 0; j < 4; ++j) {
        const int c = lane + 32 * j;
        if (c < NCLS) {
          const float e = er[c];
          if (e > 0.0f) {  // p == 0 falls in no (lower, upper] bin
            const float p = e * inv;
            int bin = (int)ceilf(p * (float)NBINS) - 1;
            bin = bin < 0 ? 0 : (bin > NBINS - 1 ? NBINS - 1 : bin);
            const int seg = c * NBINS + bin;  // stride 15 words: 32 banks
            atomicAdd(&h_conf[seg], p);
            if (c == lbl) atomicAdd(&h_acc[seg], 1.0f);
          }
        }
      }
    }
  }

  __syncthreads();
  for (int i = tid; i < HIST_SEGS; i += blockDim.x) {
    const float cv = h_conf[i];
    const float av = h_acc[i];
    if (cv != 0.0f) atomicAdd(&g_conf[i], cv);
    if (av != 0.0f) atomicAdd(&g_acc[i], av);
  }
}

__global__ void ece_finalize(const float* __restrict__ g_conf,
                             const float* __restrict__ g_acc,
                             float* __restrict__ out, int N) {
  __shared__ float s[128];
  const int c = threadIdx.x;
  float pc = 0.0f;
  if (c < NCLS) {
    float t = 0.0f;
#pragma unroll
    for (int b = 0; b < NBINS; ++b)
      t += fabsf(g_conf[c * NBINS + b] - g_acc[c * NBINS + b]);
    pc = t / (float)N;
    out[1 + c] = pc;  // per_class_sce
  }
  s[c] = pc;
  __syncthreads();
  for (int off = 64; off > 0; off >>= 1) {
    if (c < off) s[c] += s[c + off];
    __syncthreads();
  }
  if (c == 0) out[0] = s[0] / (float)NCLS;  // sce
}

extern "C" void kernel_launch(void* const* d_in, const int* in_sizes, int n_in,
                              void* d_out, int out_size, void* d_ws, size_t ws_size,
                              hipStream_t stream) {
  const float* logits = (const float*)d_in[0];          // [N, 100] fp32
  const long long* labels = (const long long*)d_in[1];  // [N] int64
  const int N = in_sizes[1];

  float* g_conf = (float*)d_ws;        // [1500]
  float* g_acc  = g_conf + HIST_SEGS;  // [1500]  (12 KB total)

  // Harness does not re-zero d_ws between replays: clear accumulators first.
  ece_zero_ws<<<4, 256, 0, stream>>>(g_conf, 2 * HIST_SEGS);
  ece_hist_kernel<<<512, 256, 0, stream>>>(logits, labels, g_conf, g_acc, N);
  ece_finalize<<<1, 128, 0, stream>>>(g_conf, g_acc, (float*)d_out, N);
}